// FilterDetections_49306224558676
// MI455X (gfx1250) — compile-verified
//
#include <hip/hip_runtime.h>
#include <hip/hip_bf16.h>
#include <stdint.h>

// ---------------------------------------------------------------------------
// FilterDetections (EfficientDet-style per-class greedy NMS + top-k gather)
// MI455X / gfx1250 design:
//  * 128 independent NMS problems -> one 1024-thread (32 x wave32) WG each.
//  * Score state in VGPRs (48/thread, fully unrolled) -> zero memory traffic
//    for the 100 serial argmax+suppress sweeps.
//  * Box stream (785KB/iter from L2) double-buffered through LDS with
//    GLOBAL_LOAD_ASYNC_TO_LDS_B128 + S_WAIT_ASYNCCNT (CDNA5 async path).
//  * Cross-wave argmax via wave32 shfl + DS_MAX_U64 on a packed (score,idx)
//    key; 3 barriers per iteration total.
//  * No matmul FLOPs exist in this workload -> no WMMA by design.
// ---------------------------------------------------------------------------

#define B_       16
#define N_       49104
#define C_       8
#define MAXDET_  100
#define NCAND    (C_ * MAXDET_)     // 800
#define SCORE_T_ 0.01f
#define IOU_T_   0.5f

#define TPB     1024                // 32 waves of 32
#define EPT     48                  // 48*1024 = 49152 >= N_
#define TILE    8192                // boxes per LDS tile (128 KB)
#define QPT     8                   // TILE / TPB async B128 issues per thread
#define NTILES  6                   // 6*8192 = 49152 covers N_
#define K1_LDS  (2 * TILE * 16 + 16) // two tiles + 8B argmax key (262,160 B)

// Monotone bijection float -> uint32 (preserves ordering incl. -inf).
__device__ __forceinline__ uint32_t fkey(float s) {
  uint32_t u = __float_as_uint(s);
  uint32_t m = (uint32_t)((int32_t)u >> 31) | 0x80000000u;
  return u ^ m;                      // fkey(-inf) == 0x007FFFFF
}

__device__ __forceinline__ unsigned long long wave_max64(unsigned long long k) {
#pragma unroll
  for (int off = 16; off > 0; off >>= 1) {
    unsigned long long o = __shfl_xor(k, off, 32);
    k = (o > k) ? o : k;
  }
  return k;
}

// ---------------------------------------------------------------------------
// Kernel 1: per-(batch,class) greedy NMS. grid = B_*C_, block = 1024.
// ---------------------------------------------------------------------------
__global__ __launch_bounds__(TPB, 1)
void nms_per_class(const float* __restrict__ boxes,
                   const float* __restrict__ cls,
                   int* __restrict__ ws_idx,
                   int* __restrict__ ws_ok) {
  extern __shared__ char smem[];
  float4* s_tiles = (float4*)smem;                                   // 2*TILE
  unsigned long long* s_key =
      (unsigned long long*)(smem + (size_t)2 * TILE * 16);

  const int tid  = threadIdx.x;
  const int b    = blockIdx.x / C_;
  const int c    = blockIdx.x % C_;
  const int pair = blockIdx.x;
  const float NEGINF = -__builtin_inff();

  const float4* boxes_b = (const float4*)boxes + (size_t)b * N_;
  const float*  cls_b   = cls + (size_t)b * N_ * C_ + c;

  // ---- load + threshold scores into a register file (thread-private) ----
  float sc[EPT];
#pragma unroll
  for (int k = 0; k < EPT; ++k) {
    int j = tid + k * TPB;
    float s = (j < N_) ? cls_b[(size_t)j * C_] : NEGINF;
    sc[k] = (s > SCORE_T_) ? s : NEGINF;
  }
  if (tid == 0) *s_key = 0ull;
  __syncthreads();

  for (int it = 0; it < MAXDET_; ++it) {
    // -- issue async DMA of tile 0 into LDS buf0; overlaps the argmax phase --
#pragma unroll
    for (int q = 0; q < QPT; ++q) {
      int j  = tid + q * TPB;
      int jj = (j < N_) ? j : (N_ - 1);
      uint32_t lds = (uint32_t)(uintptr_t)&s_tiles[tid + q * TPB];
      asm volatile("global_load_async_to_lds_b128 %0, %1, off"
                   :: "v"(lds), "v"(boxes_b + jj) : "memory");
    }

    // -- register argmax (strict > keeps lowest index, matching jnp.argmax) --
    float bv = sc[0];
    int   bj = tid;
#pragma unroll
    for (int k = 1; k < EPT; ++k) {
      int j = tid + k * TPB;
      if (sc[k] > bv) { bv = sc[k]; bj = j; }
    }
    unsigned long long key =
        ((unsigned long long)fkey(bv) << 32) |
        (unsigned long long)(0xFFFFFFFFu - (uint32_t)bj);
    key = wave_max64(key);
    if ((tid & 31) == 0) atomicMax(s_key, key);   // ds_max_u64
    __syncthreads();
    unsigned long long kk = *s_key;
    int  selIdx = (int)(0xFFFFFFFFu - (uint32_t)kk);
    bool ok     = (uint32_t)(kk >> 32) > 0x007FFFFFu;   // > fkey(-inf)
    __syncthreads();
    if (tid == 0) *s_key = 0ull;
    __syncthreads();

    if (!ok) {  // all remaining iterations would repeat this argmax
      for (int r = it + tid; r < MAXDET_; r += TPB) {
        ws_idx[pair * MAXDET_ + r] = selIdx;
        ws_ok [pair * MAXDET_ + r] = 0;
      }
      break;    // uniform; implicit wait-idle at s_endpgm drains async loads
    }
    if (tid == 0) {
      ws_idx[pair * MAXDET_ + it] = selIdx;
      ws_ok [pair * MAXDET_ + it] = 1;
    }

    // winner box: identical address across all lanes -> single L2 request
    float4 sb = boxes_b[selIdx];
    float  selArea = (sb.z - sb.x) * (sb.w - sb.y);

    // -- IoU suppression over double-buffered async box tiles --
#pragma unroll
    for (int t = 0; t < NTILES; ++t) {
      if (t + 1 < NTILES) {
        int tb  = (t + 1) * TILE;
        int buf = (t + 1) & 1;
#pragma unroll
        for (int q = 0; q < QPT; ++q) {
          int j  = tb + tid + q * TPB;
          int jj = (j < N_) ? j : (N_ - 1);
          uint32_t lds =
              (uint32_t)(uintptr_t)&s_tiles[buf * TILE + tid + q * TPB];
          asm volatile("global_load_async_to_lds_b128 %0, %1, off"
                       :: "v"(lds), "v"(boxes_b + jj) : "memory");
        }
        // 8 newer loads outstanding; async loads retire in order, so <=8
        // guarantees tile t's 8 transfers have landed in LDS.
        asm volatile("s_wait_asynccnt 0x8" ::: "memory");
      } else {
        asm volatile("s_wait_asynccnt 0x0" ::: "memory");
      }
      int buf = t & 1;
#pragma unroll
      for (int q = 0; q < QPT; ++q) {
        int k = t * QPT + q;                 // compile-time after unroll
        int j = tid + k * TPB;
        float4 bx = s_tiles[buf * TILE + tid + q * TPB];  // self-loaded slot
        float xx1 = fmaxf(sb.x, bx.x), yy1 = fmaxf(sb.y, bx.y);
        float xx2 = fminf(sb.z, bx.z), yy2 = fminf(sb.w, bx.w);
        float inter = fmaxf(xx2 - xx1, 0.f) * fmaxf(yy2 - yy1, 0.f);
        float area  = (bx.z - bx.x) * (bx.w - bx.y);
        float denom = fmaxf(selArea + area - inter, 1e-8f);
        bool  sup   = (inter > IOU_T_ * denom) || (j == selIdx);
        if ((j < N_) && sup) sc[k] = NEGINF;
      }
    }
  }
}

// ---------------------------------------------------------------------------
// Kernel 2: per-batch exact top-100 of 800 candidates + output gather.
// ---------------------------------------------------------------------------
__global__ __launch_bounds__(256, 1)
void topk_gather(const float* __restrict__ boxes,
                 const float* __restrict__ cls,
                 const float* __restrict__ rot,
                 const float* __restrict__ trans,
                 const int* __restrict__ ws_idx,
                 const int* __restrict__ ws_ok,
                 float* __restrict__ out) {
  __shared__ unsigned long long s_keys[NCAND];
  __shared__ int s_idx[NCAND];
  __shared__ int s_win[MAXDET_];
  __shared__ unsigned long long s_red;

  const int tid = threadIdx.x;
  const int b   = blockIdx.x;
  const float NEGINF = -__builtin_inff();

  for (int m = tid; m < NCAND; m += 256) {
    int c   = m / MAXDET_;
    int k   = m % MAXDET_;
    int idx = ws_idx[(b * C_ + c) * MAXDET_ + k];
    int okf = ws_ok [(b * C_ + c) * MAXDET_ + k];
    float s = okf ? cls[((size_t)b * N_ + idx) * C_ + c] : NEGINF;
    // low word ~m -> stable lowest-index tie-break, as in lax.top_k
    s_keys[m] = ((unsigned long long)fkey(s) << 32) |
                (unsigned long long)(0xFFFFFFFFu - (uint32_t)m);
    s_idx[m] = idx;
  }
  if (tid == 0) s_red = 0ull;
  __syncthreads();

  for (int r = 0; r < MAXDET_; ++r) {
    unsigned long long best = 0ull;
    for (int m = tid; m < NCAND; m += 256) {
      unsigned long long k = s_keys[m];
      best = (k > best) ? k : best;
    }
    best = wave_max64(best);
    if ((tid & 31) == 0) atomicMax(&s_red, best);
    __syncthreads();
    unsigned long long kk = s_red;
    __syncthreads();
    if (tid == 0) {
      int mw = (int)(0xFFFFFFFFu - (uint32_t)kk);
      s_win[r]   = mw;
      s_keys[mw] = 0ull;   // remove winner (any live key > 0)
      s_red      = 0ull;
    }
    __syncthreads();
  }

  if (tid < MAXDET_) {
    int  m   = s_win[tid];
    int  c   = m / MAXDET_;
    int  k   = m % MAXDET_;
    int  idx = s_idx[m];
    bool ok  = ws_ok[(b * C_ + c) * MAXDET_ + k] != 0;

    float* oB = out;                              // [B,100,4]
    float* oS = out + (size_t)B_ * MAXDET_ * 4;   // [B,100]
    float* oL = oS  + (size_t)B_ * MAXDET_;       // [B,100] labels (as f32)
    float* oR = oL  + (size_t)B_ * MAXDET_;       // [B,100,3]
    float* oT = oR  + (size_t)B_ * MAXDET_ * 3;   // [B,100,3]
    int base = b * MAXDET_ + tid;

    float4 bx = ((const float4*)boxes)[(size_t)b * N_ + idx];
    float  s  = cls[((size_t)b * N_ + idx) * C_ + c];
    const float* rp = rot   + ((size_t)b * N_ + idx) * 3;
    const float* tp = trans + ((size_t)b * N_ + idx) * 3;

    oB[base * 4 + 0] = ok ? bx.x : -1.f;
    oB[base * 4 + 1] = ok ? bx.y : -1.f;
    oB[base * 4 + 2] = ok ? bx.z : -1.f;
    oB[base * 4 + 3] = ok ? bx.w : -1.f;
    oS[base]         = ok ? s : -1.f;
    oL[base]         = ok ? (float)c : -1.f;
    oR[base * 3 + 0] = ok ? rp[0] : -1.f;
    oR[base * 3 + 1] = ok ? rp[1] : -1.f;
    oR[base * 3 + 2] = ok ? rp[2] : -1.f;
    oT[base * 3 + 0] = ok ? tp[0] : -1.f;
    oT[base * 3 + 1] = ok ? tp[1] : -1.f;
    oT[base * 3 + 2] = ok ? tp[2] : -1.f;
  }
}

extern "C" void kernel_launch(void* const* d_in, const int* in_sizes, int n_in,
                              void* d_out, int out_size, void* d_ws,
                              size_t ws_size, hipStream_t stream) {
  (void)in_sizes; (void)n_in; (void)out_size; (void)ws_size;
  const float* boxes = (const float*)d_in[0];
  const float* cls   = (const float*)d_in[1];
  const float* rot   = (const float*)d_in[2];
  const float* trans = (const float*)d_in[3];
  float* out = (float*)d_out;

  int* wsI = (int*)d_ws;                       // [B*C*100] selected indices
  int* wsO = wsI + B_ * C_ * MAXDET_;          // [B*C*100] ok flags

  // Opt in to >64KB dynamic LDS (gfx1250 WGP has 320KB). Idempotent.
  (void)hipFuncSetAttribute((const void*)nms_per_class,
                            hipFuncAttributeMaxDynamicSharedMemorySize,
                            K1_LDS);

  nms_per_class<<<B_ * C_, TPB, K1_LDS, stream>>>(boxes, cls, wsI, wsO);
  topk_gather<<<B_, 256, 0, stream>>>(boxes, cls, rot, trans, wsI, wsO, out);
}